// Net_30623116820615
// MI455X (gfx1250) — compile-verified
//
#include <hip/hip_runtime.h>
#include <hip/hip_bf16.h>

// CDNA5 (gfx1250) wave32 WMMA fp32 GEMM pipeline for the GCN layer:
//   h1 = x @ W1            [B,N,F]x[F,H]  -> ws as [N, B*H] = [4096,256]
//   agg = relu(a @ h1 +b1) [N,N]x[N,B*H]  -> ws [4096,256]   (dominant GEMM)
//   out = agg @ W2 + b2    per batch [N,H]x[H,L] -> [B,N,L]
// All matrix math via V_WMMA_F32_16X16X4_F32 (exact fp32).
// Waves own multi-tile output slabs so the streamed A fragment is reused
// across several WMMAs (A-load : WMMA ratio reduced vs 1 tile/wave).

typedef __attribute__((ext_vector_type(2))) float v2f;
typedef __attribute__((ext_vector_type(8))) float v8f;

#define NB   4096     // N nodes
#define BB   8        // batch
#define FF   240      // in features
#define HH   32       // hidden
#define LL   240      // labels
#define CC   (BB*HH)  // 256 folded columns

// ---------------------------------------------------------------------------
// Fragment layouts for V_WMMA_F32_16X16X4_F32 (wave32).
// A (16x4, MxK): lanes 0-15 -> M=lane, K={0,1} in {.x,.y}; lanes 16-31 -> M=lane-16, K={2,3}.
// B (4x16, KxN): lanes 0-15 -> N=lane, K={0,1};            lanes 16-31 -> N=lane-16, K={2,3}.
// C/D (16x16):   VGPR r -> row (r + 8*(lane>=16)), col lane%16.
// ---------------------------------------------------------------------------

__device__ __forceinline__ v2f load_b_frag(const float* __restrict__ base,
                                           int k0, int ld, int col,
                                           int l16, int half) {
    const float* p = base + (size_t)(k0 + 2 * half) * ld + col + l16;
    v2f r; r.x = p[0]; r.y = p[ld];
    return r;
}

#define WMMA_F32(c, a, b) \
    (c) = __builtin_amdgcn_wmma_f32_16x16x4_f32(false, (a), false, (b), (short)0, (c), false, false)

// ---------------------------------------------------------------------------
// Kernel 1: h1[m, b*32+h] = sum_f x[b,m,f] * W1[f,h]
// One wave per 16x32 slab (full H): tiles = B * (N/16) = 2048 waves.
// ---------------------------------------------------------------------------
__global__ void __launch_bounds__(256)
k1_feat(const float* __restrict__ x, const float* __restrict__ W1,
        float* __restrict__ h1) {
    const int wave = blockIdx.x * 8 + (threadIdx.x >> 5);
    const int lane = threadIdx.x & 31;
    const int half = lane >> 4, l16 = lane & 15;

    const int b  = wave / (NB / 16);
    const int m0 = (wave % (NB / 16)) * 16;

    const float* xrow = x + ((size_t)b * NB + m0 + l16) * FF;
    v8f acc0 = {}, acc1 = {};
#pragma unroll 5
    for (int k0 = 0; k0 < FF; k0 += 4) {
        v2f af; af.x = xrow[k0 + 2 * half]; af.y = xrow[k0 + 2 * half + 1];
        v2f b0 = load_b_frag(W1, k0, HH, 0, l16, half);
        v2f b1 = load_b_frag(W1, k0, HH, 16, l16, half);
        WMMA_F32(acc0, af, b0);
        WMMA_F32(acc1, af, b1);
    }
    float* dst = h1 + (size_t)(m0 + 8 * half) * CC + b * HH + l16;
#pragma unroll
    for (int r = 0; r < 8; ++r) {
        dst[(size_t)r * CC]      = acc0[r];
        dst[(size_t)r * CC + 16] = acc1[r];
    }
}

// ---------------------------------------------------------------------------
// Kernel 2: agg[n, c] = relu( sum_m a[n,m] * h1[m,c] + b1[c%32] )
// Single big GEMM [4096x4096]x[4096x256]; reads `a` once (L2-resident).
// One wave per 16x32 slab: (N/16) * (CC/32) = 256*8 = 2048 waves.
// A fragment (streamed 64 MB operand) reused across both column tiles.
// ---------------------------------------------------------------------------
__global__ void __launch_bounds__(256)
k2_agg(const float* __restrict__ a, const float* __restrict__ h1,
       const float* __restrict__ b1, float* __restrict__ agg) {
    const int wave = blockIdx.x * 8 + (threadIdx.x >> 5);
    const int lane = threadIdx.x & 31;
    const int half = lane >> 4, l16 = lane & 15;

    const int n0 = (wave >> 3) * 16;        // output row tile (node)
    const int c0 = (wave & 7) * 32;         // folded column slab (2 tiles)

    const float* arow = a + (size_t)(n0 + l16) * NB;   // this lane's A row
    v8f acc0 = {}, acc1 = {};
#pragma unroll 4
    for (int k0 = 0; k0 < NB; k0 += 4) {
        if ((k0 & 127) == 0 && k0 + 128 < NB)
            __builtin_prefetch(arow + k0 + 128, 0, 1);   // global_prefetch_b8
        v2f af; af.x = arow[k0 + 2 * half]; af.y = arow[k0 + 2 * half + 1];
        v2f b0 = load_b_frag(h1, k0, CC, c0, l16, half);
        v2f b1 = load_b_frag(h1, k0, CC, c0 + 16, l16, half);
        WMMA_F32(acc0, af, b0);
        WMMA_F32(acc1, af, b1);
    }
    const int col = c0 + l16;
    const float bias0 = b1[col & (HH - 1)];
    const float bias1 = b1[(col + 16) & (HH - 1)];
    float* dst = agg + (size_t)(n0 + 8 * half) * CC + col;
#pragma unroll
    for (int r = 0; r < 8; ++r) {
        float v0 = acc0[r] + bias0;
        float v1 = acc1[r] + bias1;
        dst[(size_t)r * CC]      = v0 > 0.0f ? v0 : 0.0f;
        dst[(size_t)r * CC + 16] = v1 > 0.0f ? v1 : 0.0f;
    }
}

// ---------------------------------------------------------------------------
// Kernel 3: out[b,n,l] = sum_h agg[n, b*32+h] * W2[h,l] + b2[l]
// One wave per 16x48 slab (3 L-tiles): B * (N/16) * (L/48) = 8*256*5 = 10240 waves.
// K=32 -> 8 K-steps, A fragment reused across 3 WMMAs.
// ---------------------------------------------------------------------------
__global__ void __launch_bounds__(256)
k3_head(const float* __restrict__ agg, const float* __restrict__ W2,
        const float* __restrict__ b2, float* __restrict__ out) {
    const int wave = blockIdx.x * 8 + (threadIdx.x >> 5);
    const int lane = threadIdx.x & 31;
    const int half = lane >> 4, l16 = lane & 15;

    const int slabsPerB = (NB / 16) * (LL / 48);   // 256*5
    const int b   = wave / slabsPerB;
    const int rem = wave % slabsPerB;
    const int m0  = (rem / (LL / 48)) * 16;        // node-row tile
    const int l0  = (rem % (LL / 48)) * 48;        // label-col slab

    const float* arow = agg + (size_t)(m0 + l16) * CC + b * HH;
    v8f acc0 = {}, acc1 = {}, acc2 = {};
#pragma unroll
    for (int k0 = 0; k0 < HH; k0 += 4) {
        v2f af; af.x = arow[k0 + 2 * half]; af.y = arow[k0 + 2 * half + 1];
        v2f b0 = load_b_frag(W2, k0, LL, l0, l16, half);
        v2f b1 = load_b_frag(W2, k0, LL, l0 + 16, l16, half);
        v2f b2f = load_b_frag(W2, k0, LL, l0 + 32, l16, half);
        WMMA_F32(acc0, af, b0);
        WMMA_F32(acc1, af, b1);
        WMMA_F32(acc2, af, b2f);
    }
    const float bias0 = b2[l0 + l16];
    const float bias1 = b2[l0 + 16 + l16];
    const float bias2 = b2[l0 + 32 + l16];
    float* dst = out + ((size_t)b * NB + m0 + 8 * half) * LL + l0 + l16;
#pragma unroll
    for (int r = 0; r < 8; ++r) {
        dst[(size_t)r * LL]      = acc0[r] + bias0;
        dst[(size_t)r * LL + 16] = acc1[r] + bias1;
        dst[(size_t)r * LL + 32] = acc2[r] + bias2;
    }
}

// ---------------------------------------------------------------------------
extern "C" void kernel_launch(void* const* d_in, const int* in_sizes, int n_in,
                              void* d_out, int out_size, void* d_ws, size_t ws_size,
                              hipStream_t stream) {
    const float* x  = (const float*)d_in[0];   // [B,N,F]
    const float* a  = (const float*)d_in[1];   // [N,N]
    const float* W1 = (const float*)d_in[2];   // [F,H]
    const float* b1 = (const float*)d_in[3];   // [H]
    const float* W2 = (const float*)d_in[4];   // [H,L]
    const float* b2 = (const float*)d_in[5];   // [L]
    float* out = (float*)d_out;                // [B,N,L]

    float* h1  = (float*)d_ws;                 // [N, 256] = 4 MB
    float* agg = h1 + (size_t)NB * CC;         // [N, 256] = 4 MB

    // K1: 2048 waves / 8 per block
    k1_feat<<<dim3(2048 / 8), dim3(256), 0, stream>>>(x, W1, h1);
    // K2: 2048 waves
    k2_agg<<<dim3(2048 / 8), dim3(256), 0, stream>>>(a, h1, b1, agg);
    // K3: 10240 waves
    k3_head<<<dim3(10240 / 8), dim3(256), 0, stream>>>(agg, W2, b2, out);
}